// ACOPFGeneral_39694087749650
// MI455X (gfx1250) — compile-verified
//
#include <hip/hip_runtime.h>

#define T_    4
#define N_    1024
#define E_    16384
#define H_    128
#define L_    3
#define INCH  4
#define ECH   2
#define OUTCH 4

typedef __attribute__((ext_vector_type(2))) float v2f;
typedef __attribute__((ext_vector_type(8))) float v8f;

__device__ __forceinline__ void atomicMaxF(float* addr, float val) {
    // works given init = -inf; monotone over float ordering via int/uint views
    if (val >= 0.0f) atomicMax((int*)addr, __float_as_int(val));
    else             atomicMin((unsigned int*)addr, __float_as_uint(val));
}

// ---------- per-(l,t) segment state init ----------
__global__ __launch_bounds__(256)
void k_init_seg(float* __restrict__ agg, float* __restrict__ denom,
                float* __restrict__ amax) {
    int i = blockIdx.x * blockDim.x + threadIdx.x;
    if (i < N_ * H_) agg[i] = 0.0f;
    if (i < N_) { denom[i] = 0.0f; amax[i] = -__builtin_inff(); }
}

// ---------- edge messages via V_WMMA_F32_16X16X4_F32 ----------
// One wave handles a 16-edge tile x full H=128 (8 N-blocks of 16).
// msg = xj @ Wm + bm + ea @ We + be ; alpha = leaky_relu(msg . att, 0.2)
__global__ __launch_bounds__(32)
void k_msg_alpha(const float* __restrict__ x_all,   // [T*N, 4]
                 const int*   __restrict__ src,     // [E]
                 const int*   __restrict__ dst,     // [E]
                 const float* __restrict__ ea,      // [E, 2]
                 const float* __restrict__ Wm,      // [4, 128]
                 const float* __restrict__ bm,      // [128]
                 const float* __restrict__ We,      // [2, 128]
                 const float* __restrict__ be,      // [128]
                 const float* __restrict__ attv,    // [128]
                 float* __restrict__ msg,           // [E, 128]
                 float* __restrict__ alpha,         // [E]
                 float* __restrict__ amax)          // [N]
{
    const int lane = threadIdx.x;
    const int m    = lane & 15;       // row within tile (A) / col (B,D)
    const int half = lane >> 4;       // K-half for A/B
    const int kb   = half * 2;        // K base: 0 or 2
    const int eb   = blockIdx.x * 16;
    const int e    = eb + m;

    // A-matrix (16x4 f32): lanes 0-15 hold K=0,1 ; lanes 16-31 hold K=2,3
    const int s = src[e];
    v2f a1; a1.x = x_all[s * INCH + kb]; a1.y = x_all[s * INCH + kb + 1];
    // edge-attr A (K=2 padded to 4): upper half is zero
    v2f a2;
    if (half == 0) { a2.x = ea[e * ECH + 0]; a2.y = ea[e * ECH + 1]; }
    else           { a2.x = 0.0f;            a2.y = 0.0f; }

    float ap[8];
#pragma unroll
    for (int r = 0; r < 8; ++r) ap[r] = 0.0f;

#pragma unroll
    for (int nb = 0; nb < 8; ++nb) {
        const int col = nb * 16 + m;
        // B-matrix (4x16 f32): lanes 0-15 K=0,1 ; lanes 16-31 K=2,3
        v2f b1; b1.x = Wm[(kb + 0) * H_ + col]; b1.y = Wm[(kb + 1) * H_ + col];
        v2f b2;
        if (half == 0) { b2.x = We[0 * H_ + col]; b2.y = We[1 * H_ + col]; }
        else           { b2.x = 0.0f;             b2.y = 0.0f; }

        // C init = broadcast bias (bias depends only on column n)
        const float bias = bm[col] + be[col];
        v8f c;
#pragma unroll
        for (int r = 0; r < 8; ++r) c[r] = bias;

        c = __builtin_amdgcn_wmma_f32_16x16x4_f32(false, a1, false, b1,
                                                  (short)0, c, false, false);
        c = __builtin_amdgcn_wmma_f32_16x16x4_f32(false, a2, false, b2,
                                                  (short)0, c, false, false);

        // D layout: VGPR r -> row (r + half*8), N = lane&15 (+ nb*16)
        const float av = attv[col];
#pragma unroll
        for (int r = 0; r < 8; ++r) {
            const int row = r + half * 8;
            msg[(size_t)(eb + row) * H_ + col] = c[r];
            ap[r] += c[r] * av;
        }
    }

    // reduce each ap[r] across the 16 lanes of its half (xor masks stay in-half)
#pragma unroll
    for (int mask = 1; mask < 16; mask <<= 1) {
#pragma unroll
        for (int r = 0; r < 8; ++r) ap[r] += __shfl_xor(ap[r], mask, 32);
    }

    if (m == 0) {
#pragma unroll
        for (int r = 0; r < 8; ++r) {
            const int row = r + half * 8;
            float al = ap[r];
            al = (al > 0.0f) ? al : 0.2f * al;   // leaky_relu(0.2)
            alpha[eb + row] = al;
            atomicMaxF(&amax[dst[eb + row]], al);
        }
    }
}

// ---------- softmax numerator + denominator ----------
__global__ __launch_bounds__(256)
void k_softmax_num(const int* __restrict__ dst, const float* __restrict__ amax,
                   float* __restrict__ alpha, float* __restrict__ denom) {
    int e = blockIdx.x * blockDim.x + threadIdx.x;
    if (e >= E_) return;
    int d = dst[e];
    float a = __expf(alpha[e] - amax[d]);
    alpha[e] = a;
    atomicAdd(&denom[d], a);
}

// ---------- weighted scatter-aggregate (float4 msg reads -> b128 vmem) ----------
__global__ __launch_bounds__(256)
void k_aggregate(const int* __restrict__ dst, const float* __restrict__ alpha,
                 const float* __restrict__ denom, const float* __restrict__ msg,
                 float* __restrict__ agg) {
    int tid = blockIdx.x * blockDim.x + threadIdx.x;   // over E * (H/4)
    int e  = tid >> 5;            // edge
    int hq = (tid & 31) * 4;      // hidden base (fast-varying -> coalesced b128)
    int d = dst[e];
    float w = alpha[e] / (denom[d] + 1e-16f);
    const float4 m4 = *(const float4*)(msg + (size_t)e * H_ + hq);
    float* a = agg + (size_t)d * H_ + hq;
    atomicAdd(a + 0, m4.x * w);
    atomicAdd(a + 1, m4.y * w);
    atomicAdd(a + 2, m4.z * w);
    atomicAdd(a + 3, m4.w * w);
}

// ---------- self term + L2 normalize + quarter sums -> g ----------
__global__ __launch_bounds__(128)
void k_node_out(const float* __restrict__ agg,     // [N, 128]
                const float* __restrict__ x_self,  // [N, 4]
                const float* __restrict__ Ws,      // [4, 128]
                const float* __restrict__ bs,      // [128]
                float* __restrict__ gout)          // -> g[t, l*N + n, 0..3]
{
    __shared__ float red[H_];
    const int n = blockIdx.x;
    const int h = threadIdx.x;

    float acc = agg[(size_t)n * H_ + h] + bs[h];
#pragma unroll
    for (int c = 0; c < INCH; ++c) acc += x_self[n * INCH + c] * Ws[c * H_ + h];

    red[h] = acc * acc;
    __syncthreads();
    for (int st = 64; st > 0; st >>= 1) {
        if (h < st) red[h] += red[h + st];
        __syncthreads();
    }
    float nrm = fmaxf(sqrtf(red[0]), 1e-12f);
    __syncthreads();
    red[h] = acc / nrm;
    __syncthreads();
    if (h < OUTCH) {
        float sum = 0.0f;
        for (int j = 0; j < H_ / OUTCH; ++j) sum += red[h * (H_ / OUTCH) + j];
        gout[n * OUTCH + h] = sum;
    }
}

// ---------- FC head ----------
__global__ __launch_bounds__(256)
void k_out_init(const float* __restrict__ b_fc, float* __restrict__ out) {
    int i = blockIdx.x * blockDim.x + threadIdx.x;
    if (i < T_ * N_ * OUTCH) out[i] = b_fc[i];
}

#define FC_I   (L_ * N_ * OUTCH)   // 12288
#define FC_O   (N_ * OUTCH)        // 4096
#define ITILE  256                 // split-K chunk
#define OTILE  1024                // 256 threads x 4 outputs (b128 W loads)

__global__ __launch_bounds__(256)
void k_fc(const float* __restrict__ g,     // [T, 12288]
          const float* __restrict__ W_fc,  // [T, 12288, 4096]
          float* __restrict__ out)         // [T, 4096]
{
    __shared__ float sg[ITILE];
    const int t     = blockIdx.z;
    const int obase = blockIdx.y * OTILE;
    const int ibase = blockIdx.x * ITILE;
    const int tid   = threadIdx.x;

    if (tid < ITILE) sg[tid] = g[(size_t)t * FC_I + ibase + tid];
    __syncthreads();

    const int o = obase + tid * 4;                 // 4 adjacent outputs/thread
    const float* W = W_fc + ((size_t)t * FC_I + ibase) * FC_O + o;
    float ax = 0.0f, ay = 0.0f, az = 0.0f, aw = 0.0f;
#pragma unroll 4
    for (int i = 0; i < ITILE; ++i) {
        const float4 w4 = *(const float4*)(W + (size_t)i * FC_O);  // b128, 512B/wave
        const float  gi = sg[i];
        ax += gi * w4.x; ay += gi * w4.y; az += gi * w4.z; aw += gi * w4.w;
    }
    float* op = out + (size_t)t * FC_O + o;
    atomicAdd(op + 0, ax);
    atomicAdd(op + 1, ay);
    atomicAdd(op + 2, az);
    atomicAdd(op + 3, aw);
}

extern "C" void kernel_launch(void* const* d_in, const int* in_sizes, int n_in,
                              void* d_out, int out_size, void* d_ws, size_t ws_size,
                              hipStream_t stream) {
    (void)in_sizes; (void)n_in; (void)out_size; (void)ws_size;

    const float* x         = (const float*)d_in[0];   // [T,N,4]
    const int*   src_idx   = (const int*)  d_in[1];   // [T,E]
    const int*   dst_idx   = (const int*)  d_in[2];   // [T,E]
    const float* edge_attr = (const float*)d_in[3];   // [T,E,2]
    const float* W_msg     = (const float*)d_in[4];   // [L,T,4,128]
    const float* b_msg     = (const float*)d_in[5];   // [L,T,128]
    const float* W_edge    = (const float*)d_in[6];   // [L,T,2,128]
    const float* b_edge    = (const float*)d_in[7];   // [L,T,128]
    const float* att       = (const float*)d_in[8];   // [L,T,128]
    const float* W_self    = (const float*)d_in[9];   // [L,T,4,128]
    const float* b_self    = (const float*)d_in[10];  // [L,T,128]
    const float* W_fc      = (const float*)d_in[11];  // [T,12288,4096]
    const float* b_fc      = (const float*)d_in[12];  // [T,4096]
    float* out = (float*)d_out;

    // workspace layout (floats), ~9.2 MB total, reused across (l,t)
    float* ws    = (float*)d_ws;
    float* msg   = ws;                          // E*H      = 2,097,152
    float* alpha = msg   + (size_t)E_ * H_;     // E        =    16,384
    float* amax  = alpha + E_;                  // N        =     1,024
    float* denom = amax  + N_;                  // N        =     1,024
    float* agg   = denom + N_;                  // N*H      =   131,072
    float* g     = agg   + (size_t)N_ * H_;     // T*L*N*4  =    49,152

    for (int l = 0; l < L_; ++l) {
        for (int t = 0; t < T_; ++t) {
            const int lt = l * T_ + t;
            k_init_seg<<<(N_ * H_ + 255) / 256, 256, 0, stream>>>(agg, denom, amax);
            k_msg_alpha<<<E_ / 16, 32, 0, stream>>>(
                x, src_idx + (size_t)t * E_, dst_idx + (size_t)t * E_,
                edge_attr + (size_t)t * E_ * ECH,
                W_msg  + (size_t)lt * INCH * H_, b_msg  + (size_t)lt * H_,
                W_edge + (size_t)lt * ECH  * H_, b_edge + (size_t)lt * H_,
                att    + (size_t)lt * H_,
                msg, alpha, amax);
            k_softmax_num<<<E_ / 256, 256, 0, stream>>>(
                dst_idx + (size_t)t * E_, amax, alpha, denom);
            k_aggregate<<<(E_ * (H_ / 4)) / 256, 256, 0, stream>>>(
                dst_idx + (size_t)t * E_, alpha, denom, msg, agg);
            k_node_out<<<N_, H_, 0, stream>>>(
                agg, x + (size_t)t * N_ * INCH,
                W_self + (size_t)lt * INCH * H_, b_self + (size_t)lt * H_,
                g + ((size_t)t * (L_ * N_) + (size_t)l * N_) * OUTCH);
        }
    }

    k_out_init<<<(T_ * N_ * OUTCH + 255) / 256, 256, 0, stream>>>(b_fc, out);
    k_fc<<<dim3(FC_I / ITILE, FC_O / OTILE, T_), 256, 0, stream>>>(g, W_fc, out);
}